// Matrix_optimize_79370995630851
// MI455X (gfx1250) — compile-verified
//
#include <hip/hip_runtime.h>

typedef __attribute__((ext_vector_type(2))) float v2f;
typedef __attribute__((ext_vector_type(4))) float v4f;
typedef __attribute__((ext_vector_type(8))) float v8f;

#define MAX_BASIS 1024

// ---------------------------------------------------------------------------
// Pass 1: coef[b] = W[b] / sum_c |weights[b, c]|
// One block per basis row. 128-bit nontemporal streaming loads, LDS reduce.
// ---------------------------------------------------------------------------
__global__ __launch_bounds__(256) void rowsum_coef_kernel(
    const float* __restrict__ W, const float* __restrict__ weights,
    float* __restrict__ coef, int C) {
  __shared__ float red[256];
  const int b = blockIdx.x;
  const float* row = weights + (size_t)b * (size_t)C;

  float s = 0.f;
  const int nvec = C >> 2;  // rows are 16B-aligned (C*4 % 16 == 0 for C=100000)
  const v4f* rv = (const v4f*)row;
  for (int i = threadIdx.x; i < nvec; i += 256) {
    v4f v = __builtin_nontemporal_load(&rv[i]);
    s += fabsf(v.x) + fabsf(v.y) + fabsf(v.z) + fabsf(v.w);
  }
  for (int i = (nvec << 2) + threadIdx.x; i < C; i += 256)  // tail (none here)
    s += fabsf(row[i]);

  red[threadIdx.x] = s;
  __syncthreads();
  for (int off = 128; off > 0; off >>= 1) {
    if (threadIdx.x < off) red[threadIdx.x] += red[threadIdx.x + off];
    __syncthreads();
  }
  if (threadIdx.x == 0) coef[b] = W[b] / red[0];
}

// ---------------------------------------------------------------------------
// Pass 2: out[c] = sum_b coef[b] * |weights[b, c]|  as a WMMA fp32 GEMV.
// Each wave owns 16 columns; K loop steps 4 rows per v_wmma_f32_16x16x4_f32.
//
// A (16x4 f32) layout: lane L (M=L%16); VGPR0 = K=0|K=2 (half 0|1),
//                                       VGPR1 = K=1|K=3. Only row M=0 is live.
// B (4x16 f32) layout: lane L (N=L%16); VGPR0 = K=0|K=2, VGPR1 = K=1|K=3.
// D (16x16 f32):       VGPR0, lanes 0-15 = row M=0, N=0..15  -> the outputs.
//
// All masking is arithmetic (mul by {0,1}) so the k-loop has no divergence:
// loads stay unconditional (full MLP) and EXEC is all-1s at every WMMA.
// ---------------------------------------------------------------------------
__global__ __launch_bounds__(256) void wmma_gemv_kernel(
    const float* __restrict__ weights, const float* __restrict__ coef,
    float* __restrict__ out, int B, int C) {
  __shared__ float lcoef[MAX_BASIS];
  for (int i = threadIdx.x; i < B; i += 256) lcoef[i] = coef[i];
  __syncthreads();

  const int wave = threadIdx.x >> 5;         // wave32
  const int lane = threadIdx.x & 31;
  const int col0 = (blockIdx.x * 8 + wave) * 16;
  if (col0 >= C) return;                      // uniform per wave

  const int half = lane >> 4;                 // 0: lanes 0-15, 1: lanes 16-31
  const int n = lane & 15;
  const float amsk = (n == 0) ? 1.f : 0.f;    // lanes holding A row M=0
  const float bmsk = (col0 + n) < C ? 1.f : 0.f;
  const int colc = min(col0 + n, C - 1);      // always-in-bounds column

  const size_t rowC = (size_t)C;
  const float* p = weights + (size_t)(2 * half) * rowC + (size_t)colc;

  v8f acc = {0.f, 0.f, 0.f, 0.f, 0.f, 0.f, 0.f, 0.f};

#pragma unroll 2
  for (int k0 = 0; k0 < B; k0 += 4) {
    // coef pair for this half-wave's K rows: one aligned 64-bit LDS read,
    // then zeroed in all lanes except M=0 holders (lane 0 / lane 16).
    v2f c = *(const v2f*)&lcoef[k0 + 2 * half];
    v2f a;
    a.x = c.x * amsk;
    a.y = c.y * amsk;

    v2f bm;
    bm.x = fabsf(__builtin_nontemporal_load(p)) * bmsk;         // row ka
    bm.y = fabsf(__builtin_nontemporal_load(p + rowC)) * bmsk;  // row ka+1

    acc = __builtin_amdgcn_wmma_f32_16x16x4_f32(
        /*neg_a=*/false, a, /*neg_b=*/false, bm,
        /*c_mod=*/(short)0, acc, /*reuse_a=*/false, /*reuse_b=*/false);

    p += 4 * rowC;
  }

  if (lane < 16 && (col0 + lane) < C) out[col0 + lane] = acc[0];
}

// ---------------------------------------------------------------------------
extern "C" void kernel_launch(void* const* d_in, const int* in_sizes, int n_in,
                              void* d_out, int out_size, void* d_ws, size_t ws_size,
                              hipStream_t stream) {
  const float* W = (const float*)d_in[0];        // (B,)
  const float* weights = (const float*)d_in[1];  // (B, C) row-major
  const int B = in_sizes[0];                     // 1024
  const int C = in_sizes[1] / B;                 // 100000
  float* coef = (float*)d_ws;                    // B floats of scratch
  float* out = (float*)d_out;                    // (C, 1) flat = C floats

  rowsum_coef_kernel<<<B, 256, 0, stream>>>(W, weights, coef, C);

  const int tiles = (C + 15) / 16;               // 16 columns per wave
  const int blocks = (tiles + 7) / 8;            // 8 waves per block
  wmma_gemv_kernel<<<blocks, 256, 0, stream>>>(weights, coef, out, B, C);
}